// DelayNeuralODEGRN_19447611916354
// MI455X (gfx1250) — compile-verified
//
#include <hip/hip_runtime.h>
#include <hip/hip_bf16.h>
#include <math.h>

// ---------------- problem constants ----------------
#define N_TFS     1024
#define N_TARGETS 16384
#define LATENT    64
#define HIDDEN    256
#define RANK      16
#define COND_DIM  16
#define TIME_DIM  13
#define OBS       (2*N_TFS + N_TARGETS)        // 18432
#define STATE_N   (OBS + LATENT)               // 18496
#define TF_IN     (2*N_TFS + LATENT + COND_DIM + TIME_DIM)   // 2141
#define BASAL_IN  (N_TARGETS + LATENT + COND_DIM + TIME_DIM) // 16477
#define LATENT_IN (OBS + LATENT + COND_DIM + TIME_DIM)       // 18525
#define CTX_IN    (LATENT + COND_DIM + TIME_DIM)             // 93
#define EW_ELEMS  (N_TARGETS * N_TFS)          // 16777216

// matvec tiling
#define MV_TN 1024   // columns per block (256 threads * float4)
#define MV_TK 128    // K rows per block

typedef __attribute__((ext_vector_type(2))) float v2f;
typedef __attribute__((ext_vector_type(4))) float v4f;
typedef __attribute__((ext_vector_type(8))) float v8f;

// ---------------- tiny prep kernels ----------------
__global__ void k_prep(const float* __restrict__ t_in, const int* __restrict__ cid,
                       const float* __restrict__ cond_table,
                       float* __restrict__ tfeat, float* __restrict__ cond) {
  int i = threadIdx.x;
  float t = t_in[0];
  if (i == 0) tfeat[0] = t;
  if (i >= 1 && i < TIME_DIM) {
    int f = (i - 1) >> 1;
    float ang = exp2f((float)f) * 3.14159265358979323846f * t;
    tfeat[i] = ((i - 1) & 1) ? __cosf(ang) : __sinf(ang);
  }
  if (i < COND_DIM) cond[i] = cond_table[cid[0] * COND_DIM + i];
}

__global__ void k_concat6(float* __restrict__ dst,
                          const float* s0, int n0, const float* s1, int n1,
                          const float* s2, int n2, const float* s3, int n3,
                          const float* s4, int n4, const float* s5, int n5) {
  int i = blockIdx.x * blockDim.x + threadIdx.x;
  int total = n0 + n1 + n2 + n3 + n4 + n5;
  if (i >= total) return;
  int j = i;
  if (j < n0) { dst[i] = s0[j]; return; } j -= n0;
  if (j < n1) { dst[i] = s1[j]; return; } j -= n1;
  if (j < n2) { dst[i] = s2[j]; return; } j -= n2;
  if (j < n3) { dst[i] = s3[j]; return; } j -= n3;
  if (j < n4) { dst[i] = s4[j]; return; } j -= n4;
  dst[i] = s5[j];
}

__global__ void k_copy(float* __restrict__ dst, const float* __restrict__ src, int n) {
  int i = blockIdx.x * blockDim.x + threadIdx.x;
  if (i < n) dst[i] = src[i];
}

__global__ void k_silu_ip(float* __restrict__ a, int n) {
  int i = blockIdx.x * blockDim.x + threadIdx.x;
  if (i < n) { float v = a[i]; a[i] = v / (1.0f + __expf(-v)); }
}

__global__ void k_sigmoid(float* __restrict__ dst, const float* __restrict__ a, int n) {
  int i = blockIdx.x * blockDim.x + threadIdx.x;
  if (i < n) dst[i] = 1.0f / (1.0f + __expf(-a[i]));
}

// ---------------- K-split matvec: acc[n] += sum_k x[k]*W[k*N+n] ----------------
// NT=true streams W with non-temporal loads (one-shot 268 MB matrix must not
// evict the ~94 MB of weights/activations we want resident in the 192 MB L2
// across graph replays).
template <bool NT>
__global__ void __launch_bounds__(256)
k_matvec(const float* __restrict__ x, const float* __restrict__ W,
         float* __restrict__ acc, int K, int N) {
  __shared__ float xs[MV_TK];
  const int n0 = blockIdx.x * MV_TN;
  const int k0 = blockIdx.y * MV_TK;
  int kend = K - k0; if (kend > MV_TK) kend = MV_TK;
  for (int i = threadIdx.x; i < kend; i += blockDim.x) xs[i] = x[k0 + i];
  __syncthreads();
  const int n = n0 + threadIdx.x * 4;
  if (n >= N) return;
  v4f s = {0.f, 0.f, 0.f, 0.f};
  const float* Wp = W + (size_t)k0 * (size_t)N + n;
  #pragma unroll 4
  for (int k = 0; k < kend; ++k) {
    v4f w = NT ? __builtin_nontemporal_load((const v4f*)Wp) : *(const v4f*)Wp;
    float xv = xs[k];
    s.x = fmaf(xv, w.x, s.x);
    s.y = fmaf(xv, w.y, s.y);
    s.z = fmaf(xv, w.z, s.z);
    s.w = fmaf(xv, w.w, s.w);
    Wp += N;
  }
  atomicAdd(&acc[n + 0], s.x);
  atomicAdd(&acc[n + 1], s.y);
  atomicAdd(&acc[n + 2], s.z);
  atomicAdd(&acc[n + 3], s.w);
}

// ---------------- edge_w = 0.25 * u @ v^T  (WMMA f32 16x16x4), fused with edge_w @ gate ----------------
// One wave per 16-row stripe of edge_w. A (u) fragments stay resident; 4 chained
// K=4 WMMAs per 16x16 tile, looped over all 64 column tiles. edge_w is written
// with non-temporal stores (67 MB, never re-read -- the @gate matvec is fused
// in-register, so don't let the output stream write-allocate L2).
__global__ void __launch_bounds__(32)
k_edgew(const float* __restrict__ u, const float* __restrict__ v,
        const float* __restrict__ gate, float* __restrict__ ew,
        float* __restrict__ edge_acc) {
  const int tm   = blockIdx.x;         // 0..1023 (M tile)
  const int lane = threadIdx.x;        // 0..31
  const int hf   = lane >> 4;          // half-wave select
  const int l15  = lane & 15;
  const int i0   = tm * 16;

  // A fragments per ISA 32-bit 16x4 layout: lanes 0-15 -> M rows,
  // VGPR0 = K{0 | 2}, VGPR1 = K{1 | 3} depending on half-wave.
  v2f A[4];
  {
    const float* up = u + (size_t)(i0 + l15) * RANK + hf * 2;
    #pragma unroll
    for (int kc = 0; kc < 4; ++kc) A[kc] = *(const v2f*)(up + kc * 4);
  }

  float acc[8];
  #pragma unroll
  for (int r = 0; r < 8; ++r) acc[r] = 0.f;

  const int col = l15;  // within-tile column for this lane
  for (int tn = 0; tn < 64; ++tn) {
    const int j0 = tn * 16;
    // B[k, j] = v[j, k] — same striping as A but over N.
    const float* vp = v + (size_t)(j0 + l15) * RANK + hf * 2;
    v8f C = {};
    #pragma unroll
    for (int kc = 0; kc < 4; ++kc) {
      v2f B = *(const v2f*)(vp + kc * 4);
      C = __builtin_amdgcn_wmma_f32_16x16x4_f32(
            /*neg_a=*/false, A[kc], /*neg_b=*/false, B,
            /*c_mod=*/(short)0, C, /*reuse_a=*/false, /*reuse_b=*/false);
    }
    const int c = j0 + col;
    const float g = gate[c];
    // D layout: VGPR r -> row (r + hf*8), lane -> column.
    float* out = ew + (size_t)(i0 + hf * 8) * N_TFS + c;
    #pragma unroll
    for (int r = 0; r < 8; ++r) {
      float d = C[r] * 0.25f;              // 1/sqrt(RANK)
      __builtin_nontemporal_store(d, out + (size_t)r * N_TFS);
      acc[r] = fmaf(d, g, acc[r]);
    }
  }

  // reduce across the 16 lanes of each half-wave -> per-row dot(edge_w_row, gate)
  #pragma unroll
  for (int r = 0; r < 8; ++r) {
    float s = acc[r];
    for (int off = 1; off < 16; off <<= 1) s += __shfl_xor(s, off, 32);
    if (l15 == 0) edge_acc[i0 + hf * 8 + r] = s;
  }
}

// ---------------- final assembly: dy ----------------
__global__ void k_assemble(float* __restrict__ dy, const float* __restrict__ state,
                           const float* __restrict__ tf_out, const float* __restrict__ basal_out,
                           const float* __restrict__ edge_acc, const float* __restrict__ lat_out,
                           const float* __restrict__ rdo, const float* __restrict__ rdl) {
  int i = blockIdx.x * blockDim.x + threadIdx.x;
  if (i >= STATE_N) return;
  float drive, sp;
  if (i < 2 * N_TFS) {
    sp = log1pf(__expf(rdo[i]));
    drive = tf_out[i];
  } else if (i < OBS) {
    sp = log1pf(__expf(rdo[i]));
    drive = basal_out[i - 2 * N_TFS] + edge_acc[i - 2 * N_TFS];
  } else {
    sp = log1pf(__expf(rdl[i - OBS]));
    drive = lat_out[i - OBS];
  }
  dy[i] = drive - sp * state[i];
}

// ---------------- host side ----------------
static inline int cdiv(int a, int b) { return (a + b - 1) / b; }

enum { ACT_LIN = 0, ACT_SILU = 1 };

static void layer(const float* x, const float* W, const float* b, float* acc,
                  int K, int N, int act, hipStream_t s, bool nt = false) {
  k_copy<<<cdiv(N, 256), 256, 0, s>>>(acc, b, N);
  dim3 g(cdiv(N, MV_TN), cdiv(K, MV_TK));
  if (nt) k_matvec<true><<<g, 256, 0, s>>>(x, W, acc, K, N);
  else    k_matvec<false><<<g, 256, 0, s>>>(x, W, acc, K, N);
  if (act == ACT_SILU) k_silu_ip<<<cdiv(N, 256), 256, 0, s>>>(acc, N);
}

extern "C" void kernel_launch(void* const* d_in, const int* in_sizes, int n_in,
                              void* d_out, int out_size, void* d_ws, size_t ws_size,
                              hipStream_t stream) {
  (void)in_sizes; (void)n_in; (void)out_size; (void)ws_size;
  const float* t_in   = (const float*)d_in[0];
  const float* state  = (const float*)d_in[1];
  const float* lagA   = (const float*)d_in[2];
  const float* lagT   = (const float*)d_in[3];
  const int*   cid    = (const int*)d_in[4];
  const float* ctab   = (const float*)d_in[5];
  const float* tfW1 = (const float*)d_in[6],  *tfb1 = (const float*)d_in[7];
  const float* tfW2 = (const float*)d_in[8],  *tfb2 = (const float*)d_in[9];
  const float* tfW3 = (const float*)d_in[10], *tfb3 = (const float*)d_in[11];
  const float* gW1  = (const float*)d_in[12], *gb1  = (const float*)d_in[13];
  const float* gW2  = (const float*)d_in[14], *gb2  = (const float*)d_in[15];
  const float* gW3  = (const float*)d_in[16], *gb3  = (const float*)d_in[17];
  const float* bW1  = (const float*)d_in[18], *bb1  = (const float*)d_in[19];
  const float* bW2  = (const float*)d_in[20], *bb2  = (const float*)d_in[21];
  const float* bW3  = (const float*)d_in[22], *bb3  = (const float*)d_in[23];
  const float* lW1  = (const float*)d_in[24], *lb1  = (const float*)d_in[25];
  const float* lW2  = (const float*)d_in[26], *lb2  = (const float*)d_in[27];
  const float* lW3  = (const float*)d_in[28], *lb3  = (const float*)d_in[29];
  const float* cW1  = (const float*)d_in[30], *cb1  = (const float*)d_in[31];
  const float* cW2  = (const float*)d_in[32], *cb2  = (const float*)d_in[33];
  const float* euW  = (const float*)d_in[34], *eub  = (const float*)d_in[35];
  const float* evW  = (const float*)d_in[36], *evb  = (const float*)d_in[37];
  const float* rdo  = (const float*)d_in[38];
  const float* rdl  = (const float*)d_in[39];

  // workspace layout (floats; all offsets multiple of 4 -> 16B aligned)
  float* ws = (float*)d_ws;
  float* tfeat    = ws + 0;        // 16
  float* cond     = ws + 16;       // 16
  float* in_tf    = ws + 32;       // 2144
  float* in_gate  = ws + 2176;     // 2144
  float* in_basal = ws + 4320;     // 16480
  float* in_lat   = ws + 20800;    // 18528
  float* in_ctx   = ws + 39328;    // 96
  float* h1       = ws + 39424;    // 256
  float* h2       = ws + 39680;    // 256
  float* tf_out   = ws + 39936;    // 2048
  float* gacc     = ws + 41984;    // 1024
  float* basal_o  = ws + 43008;    // 16384
  float* lat_o    = ws + 59392;    // 64
  float* ctx      = ws + 59456;    // 256
  float* u        = ws + 59712;    // 262144
  float* v        = ws + 321856;   // 16384
  float* eacc     = ws + 338240;   // 16384

  float* dy   = (float*)d_out;
  float* ew   = dy + STATE_N;                 // 16384x1024
  float* gate = dy + STATE_N + EW_ELEMS;      // 1024

  const float* a_ptr = state;                 // state[0:1024]
  const float* r_ptr = state + N_TFS;         // state[1024:2048]
  const float* x_ptr = state + 2 * N_TFS;     // state[2048:18432]
  const float* z_ptr = state + OBS;           // state[18432:18496]

  // --- prep ---
  k_prep<<<1, 32, 0, stream>>>(t_in, cid, ctab, tfeat, cond);
  k_concat6<<<cdiv(TF_IN, 256), 256, 0, stream>>>(in_tf,
      lagA, N_TFS, r_ptr, N_TFS, z_ptr, LATENT, cond, COND_DIM, tfeat, TIME_DIM, nullptr, 0);
  k_concat6<<<cdiv(TF_IN, 256), 256, 0, stream>>>(in_gate,
      lagT, N_TFS, lagT + N_TFS, N_TFS, z_ptr, LATENT, cond, COND_DIM, tfeat, TIME_DIM, nullptr, 0);
  k_concat6<<<cdiv(BASAL_IN, 256), 256, 0, stream>>>(in_basal,
      x_ptr, N_TARGETS, z_ptr, LATENT, cond, COND_DIM, tfeat, TIME_DIM, nullptr, 0, nullptr, 0);
  k_concat6<<<cdiv(LATENT_IN, 256), 256, 0, stream>>>(in_lat,
      a_ptr, N_TFS, r_ptr, N_TFS, x_ptr, N_TARGETS, z_ptr, LATENT, cond, COND_DIM, tfeat, TIME_DIM);
  k_concat6<<<cdiv(CTX_IN, 256), 256, 0, stream>>>(in_ctx,
      z_ptr, LATENT, cond, COND_DIM, tfeat, TIME_DIM, nullptr, 0, nullptr, 0, nullptr, 0);

  // --- tf_dyn MLP ---
  layer(in_tf, tfW1, tfb1, h1, TF_IN, HIDDEN, ACT_SILU, stream);
  layer(h1, tfW2, tfb2, h2, HIDDEN, HIDDEN, ACT_SILU, stream);
  layer(h2, tfW3, tfb3, tf_out, HIDDEN, 2 * N_TFS, ACT_LIN, stream);
  // --- gate MLP (+sigmoid straight into d_out) ---
  layer(in_gate, gW1, gb1, h1, TF_IN, HIDDEN, ACT_SILU, stream);
  layer(h1, gW2, gb2, h2, HIDDEN, HIDDEN, ACT_SILU, stream);
  layer(h2, gW3, gb3, gacc, HIDDEN, N_TFS, ACT_LIN, stream);
  k_sigmoid<<<cdiv(N_TFS, 256), 256, 0, stream>>>(gate, gacc, N_TFS);
  // --- basal MLP ---
  layer(in_basal, bW1, bb1, h1, BASAL_IN, HIDDEN, ACT_SILU, stream);
  layer(h1, bW2, bb2, h2, HIDDEN, HIDDEN, ACT_SILU, stream);
  layer(h2, bW3, bb3, basal_o, HIDDEN, N_TARGETS, ACT_LIN, stream);
  // --- latent MLP ---
  layer(in_lat, lW1, lb1, h1, LATENT_IN, HIDDEN, ACT_SILU, stream);
  layer(h1, lW2, lb2, h2, HIDDEN, HIDDEN, ACT_SILU, stream);
  layer(h2, lW3, lb3, lat_o, HIDDEN, LATENT, ACT_LIN, stream);
  // --- edge context MLP ---
  layer(in_ctx, cW1, cb1, h1, CTX_IN, HIDDEN, ACT_SILU, stream);
  layer(h1, cW2, cb2, ctx, HIDDEN, HIDDEN, ACT_LIN, stream);
  // --- u (268 MB one-shot stream: non-temporal), v (17 MB, keep L2-resident) ---
  layer(ctx, euW, eub, u, HIDDEN, N_TARGETS * RANK, ACT_LIN, stream, /*nt=*/true);
  layer(ctx, evW, evb, v, HIDDEN, N_TFS * RANK, ACT_LIN, stream);

  // --- WMMA: edge_w tiles + fused edge_w @ gate ---
  k_edgew<<<N_TARGETS / 16, 32, 0, stream>>>(u, v, gate, ew, eacc);

  // --- dy ---
  k_assemble<<<cdiv(STATE_N, 256), 256, 0, stream>>>(dy, state, tf_out, basal_o,
                                                     eacc, lat_o, rdo, rdl);
}